// Net_F_78881369358760
// MI455X (gfx1250) — compile-verified
//
#include <hip/hip_runtime.h>
#include <math.h>

typedef __attribute__((ext_vector_type(2))) float v2f;
typedef __attribute__((ext_vector_type(4))) float v4f;
typedef __attribute__((ext_vector_type(8))) float v8f;

#define NU_F 0.0031830988618379067f  // 0.01 / pi

constexpr int WSTR = 68;          // padded row stride (floats) for weights in LDS
constexpr int SSTR = 68;          // padded column stride (floats) for staging
constexpr int WAVES = 8;
constexpr int PTS_PER_WAVE = 16;
constexpr int PTS_PER_BLOCK = WAVES * PTS_PER_WAVE;   // 128

__device__ __forceinline__ float fast_tanh(float z) {
#if __has_builtin(__builtin_amdgcn_tanhf)
    return __builtin_amdgcn_tanhf(z);            // v_tanh_f32 (CDNA5 TRANS op)
#else
    const float e = __expf(2.0f * z);
#if __has_builtin(__builtin_amdgcn_rcpf)
    return 1.0f - 2.0f * __builtin_amdgcn_rcpf(e + 1.0f);
#else
    return 1.0f - 2.0f / (e + 1.0f);
#endif
#endif
}

// One LDS object -> member offsets are guaranteed by layout, sW lands at offset 0
// so all A-fragment accesses are (one base VGPR + imm16) DS loads.
struct __align__(16) Smem {
    float sW[3 * 64 * WSTR];                 // [0, 52224): W1..W3 padded row-major
    float sBias[3 * 64];                     // b1..b3
    float sW0[128];                          // W0 row-major (64x2)
    float sb0[64];
    float sW4[64];
    float sStage[WAVES][2][16 * SSTR];       // per-wave B staging, 2 quantities
};

__global__ __launch_bounds__(256, 1)
void pinn_burgers_residual(
    const float* __restrict__ x,  const float* __restrict__ t,
    const float* __restrict__ W0, const float* __restrict__ b0,
    const float* __restrict__ W1, const float* __restrict__ b1,
    const float* __restrict__ W2, const float* __restrict__ b2,
    const float* __restrict__ W3, const float* __restrict__ b3,
    const float* __restrict__ W4, const float* __restrict__ b4,
    float* __restrict__ out, int n)
{
    __shared__ Smem sm;

    const int tid = threadIdx.x;

    // ---- cooperative weight load (once) ----
    for (int i = tid; i < 64 * 64; i += 256) {
        const int r = i >> 6, c = i & 63;
        sm.sW[0 * 64 * WSTR + r * WSTR + c] = W1[i];
        sm.sW[1 * 64 * WSTR + r * WSTR + c] = W2[i];
        sm.sW[2 * 64 * WSTR + r * WSTR + c] = W3[i];
    }
    if (tid < 64) {
        sm.sBias[tid]       = b1[tid];
        sm.sBias[64 + tid]  = b2[tid];
        sm.sBias[128 + tid] = b3[tid];
        sm.sb0[tid] = b0[tid];
        sm.sW4[tid] = W4[tid];
    } else if (tid < 192) {
        sm.sW0[tid - 64] = W0[tid - 64];
    }
    __syncthreads();

    const int wave = tid >> 5;
    const int lane = tid & 31;
    const int half = lane >> 4;   // C-layout half: 0 -> rows g, 1 -> rows g+8
    const int lcol = lane & 15;   // point column within the wave tile

    const int p  = blockIdx.x * PTS_PER_BLOCK + wave * PTS_PER_WAVE + lcol;
    const int pc = p < n ? p : (n - 1);
    const float xv = x[pc];
    const float tv = t[pc];
    const float b4v = b4[0];

    // ---- lane-variant base pointers (computed once; hot DS accesses = base + imm16) ----
    const float* __restrict__ Ap  = &sm.sW[lcol * WSTR + 2 * half];              // A fragments
    const float* __restrict__ Bp0 = &sm.sStage[wave][0][lcol * SSTR + 2 * half]; // B fragments
    const float* __restrict__ Bp1 = &sm.sStage[wave][1][lcol * SSTR + 2 * half];
    float* __restrict__ Sp0 = &sm.sStage[wave][0][lcol * SSTR + 8 * half];       // staging stores
    float* __restrict__ Sp1 = &sm.sStage[wave][1][lcol * SSTR + 8 * half];
    const float* __restrict__ w0p = &sm.sW0[16 * half];
    const float* __restrict__ b0p = &sm.sb0[8 * half];
    const float* __restrict__ w4p = &sm.sW4[8 * half];

    const v8f vzero = {0.f, 0.f, 0.f, 0.f, 0.f, 0.f, 0.f, 0.f};

    // reg[q][mt]: q = {h, h_x, h_t, h_xx}; mt = 16-row tile of 64 neurons (WMMA C-layout)
    v8f reg[4][4];

    // ---------------- layer 0 (2 -> 64), pure VALU ----------------
#pragma unroll
    for (int mt = 0; mt < 4; ++mt) {
        v8f h, hx, ht2, hxx;
#pragma unroll
        for (int g = 0; g < 8; ++g) {
            const v2f w01 = *(const v2f*)(w0p + (mt * 16 + g) * 2);  // {wx, wt}
            const float wx = w01.x, wt = w01.y;
            const float z  = fmaf(wx, xv, fmaf(wt, tv, b0p[mt * 16 + g]));
            const float hh = fast_tanh(z);
            const float s  = 1.0f - hh * hh;
            const float ax = s * wx;
            h[g]   = hh;
            hx[g]  = ax;                      // z_x = wx
            ht2[g] = s * wt;                  // z_t = wt
            hxx[g] = -2.0f * hh * ax * wx;    // z_xx = 0
        }
        reg[0][mt] = h; reg[1][mt] = hx; reg[2][mt] = ht2; reg[3][mt] = hxx;
    }

    // ---------------- hidden layers 1..3 (64 -> 64), WMMA ----------------
#pragma unroll
    for (int layer = 0; layer < 3; ++layer) {
#pragma unroll
        for (int pair = 0; pair < 2; ++pair) {
            const int q0 = 2 * pair, q1 = 2 * pair + 1;

            // stage both quantities to LDS (column-major, B-fragment-friendly)
#pragma unroll
            for (int mt = 0; mt < 4; ++mt) {
                const v8f a = reg[q0][mt];
                const v8f b = reg[q1][mt];
                *(v4f*)(Sp0 + mt * 16)     = __builtin_shufflevector(a, a, 0, 1, 2, 3);
                *(v4f*)(Sp0 + mt * 16 + 4) = __builtin_shufflevector(a, a, 4, 5, 6, 7);
                *(v4f*)(Sp1 + mt * 16)     = __builtin_shufflevector(b, b, 0, 1, 2, 3);
                *(v4f*)(Sp1 + mt * 16 + 4) = __builtin_shufflevector(b, b, 4, 5, 6, 7);
            }
            // Same-wave LDS ops execute in order; keep IR ordering only.
            asm volatile("" ::: "memory");

            // H regs for this pair are dead (staged) -> accumulate Z in place
#pragma unroll
            for (int mt = 0; mt < 4; ++mt) { reg[q0][mt] = vzero; reg[q1][mt] = vzero; }

            // Z(64x16) = W(64x64) * H(64x16), software-pipelined:
            // issue the 6 fragment loads for ks+1 before the 8 WMMAs of ks.
            v2f af[4], bf0, bf1;
            bf0 = *(const v2f*)(Bp0);
            bf1 = *(const v2f*)(Bp1);
#pragma unroll
            for (int mt = 0; mt < 4; ++mt)
                af[mt] = *(const v2f*)(Ap + (layer * 64 + mt * 16) * WSTR);

#pragma unroll
            for (int ks = 0; ks < 16; ++ks) {
                const int kn = (ks < 15) ? (ks + 1) : 15;
                const v2f nbf0 = *(const v2f*)(Bp0 + kn * 4);
                const v2f nbf1 = *(const v2f*)(Bp1 + kn * 4);
                v2f naf[4];
#pragma unroll
                for (int mt = 0; mt < 4; ++mt)
                    naf[mt] = *(const v2f*)(Ap + (layer * 64 + mt * 16) * WSTR + kn * 4);

#pragma unroll
                for (int mt = 0; mt < 4; ++mt) {
                    reg[q0][mt] = __builtin_amdgcn_wmma_f32_16x16x4_f32(
                        false, af[mt], false, bf0, (short)0, reg[q0][mt], false, false);
                    reg[q1][mt] = __builtin_amdgcn_wmma_f32_16x16x4_f32(
                        false, af[mt], false, bf1, (short)0, reg[q1][mt], false, false);
                }
#pragma unroll
                for (int mt = 0; mt < 4; ++mt) af[mt] = naf[mt];
                bf0 = nbf0; bf1 = nbf1;
            }
            asm volatile("" ::: "memory");  // keep B reads ordered before next pair's stores
        }

        // pointwise tanh jet update (reg now holds Z, ZX, ZT, ZXX)
        const float* __restrict__ bp = &sm.sBias[layer * 64 + 8 * half];
#pragma unroll
        for (int mt = 0; mt < 4; ++mt) {
            const v8f z = reg[0][mt], zx = reg[1][mt], zt = reg[2][mt], zxx = reg[3][mt];
            v8f h, hx, ht2, hxx;
#pragma unroll
            for (int g = 0; g < 8; ++g) {
                const float zz = z[g] + bp[mt * 16 + g];   // bias only on the value path
                const float hh = fast_tanh(zz);
                const float s  = 1.0f - hh * hh;
                const float ax = s * zx[g];
                h[g]   = hh;
                hx[g]  = ax;
                ht2[g] = s * zt[g];
                hxx[g] = s * zxx[g] - 2.0f * hh * ax * zx[g];
            }
            reg[0][mt] = h; reg[1][mt] = hx; reg[2][mt] = ht2; reg[3][mt] = hxx;
        }
    }

    // ---------------- output layer (64 -> 1), VALU dot + half-wave reduce ----------------
    float u = 0.f, ux = 0.f, ut = 0.f, uxx = 0.f;
#pragma unroll
    for (int mt = 0; mt < 4; ++mt) {
#pragma unroll
        for (int g = 0; g < 8; ++g) {
            const float w = w4p[mt * 16 + g];
            u   = fmaf(w, reg[0][mt][g], u);
            ux  = fmaf(w, reg[1][mt][g], ux);
            ut  = fmaf(w, reg[2][mt][g], ut);
            uxx = fmaf(w, reg[3][mt][g], uxx);
        }
    }
    u   += __shfl_xor(u,   16, 32);
    ux  += __shfl_xor(ux,  16, 32);
    ut  += __shfl_xor(ut,  16, 32);
    uxx += __shfl_xor(uxx, 16, 32);
    u += b4v;

    if (half == 0 && p < n) {
        out[p] = ut + u * ux - NU_F * uxx;
    }
}

extern "C" void kernel_launch(void* const* d_in, const int* in_sizes, int n_in,
                              void* d_out, int out_size, void* d_ws, size_t ws_size,
                              hipStream_t stream) {
    const float* x  = (const float*)d_in[0];
    const float* t  = (const float*)d_in[1];
    const float* W0 = (const float*)d_in[2];
    const float* b0 = (const float*)d_in[3];
    const float* W1 = (const float*)d_in[4];
    const float* b1 = (const float*)d_in[5];
    const float* W2 = (const float*)d_in[6];
    const float* b2 = (const float*)d_in[7];
    const float* W3 = (const float*)d_in[8];
    const float* b3 = (const float*)d_in[9];
    const float* W4 = (const float*)d_in[10];
    const float* b4 = (const float*)d_in[11];
    float* out = (float*)d_out;

    const int n = in_sizes[0];
    const int blocks = (n + PTS_PER_BLOCK - 1) / PTS_PER_BLOCK;
    pinn_burgers_residual<<<blocks, 256, 0, stream>>>(
        x, t, W0, b0, W1, b1, W2, b2, W3, b3, W4, b4, out, n);
}